// SelfAttention_87402584474126
// MI455X (gfx1250) — compile-verified
//
#include <hip/hip_runtime.h>
#include <cmath>

// CDNA5 / gfx1250, wave32.  fp32 matmuls via V_WMMA_F32_16X16X4_F32.
// LDS-staged GEMM: async global->LDS tile copies (ASYNCcnt path), double-buffered,
// 8 waves / 256-thread block (2x4), each wave owns a 32x32 C tile
// (4 independent WMMA accumulator chains).

typedef __attribute__((ext_vector_type(2))) float v2f;
typedef __attribute__((ext_vector_type(8))) float v8f;

#define WMMA_F32(A_, B_, C_) \
  __builtin_amdgcn_wmma_f32_16x16x4_f32(false, (A_), false, (B_), (short)0, (C_), false, false)

// 32-bit LDS byte address of a generic pointer that points into LDS.
__device__ __forceinline__ unsigned lds_addr_of(const void* p) {
  return (unsigned)(unsigned long long)(__attribute__((address_space(3))) const char*)p;
}

// Per-lane async copy of 16 bytes global -> LDS (tracked by ASYNCcnt).
__device__ __forceinline__ void async_copy_b128(unsigned lds, const float* g) {
  asm volatile("global_load_async_to_lds_b128 %0, %1, off"
               :: "v"(lds), "v"(g)
               : "memory");
}

__device__ __forceinline__ void wait_async0() {
  asm volatile("s_wait_asynccnt 0" ::: "memory");
}

#define KC 32          // K chunk staged per iteration
#define AS_STR 36      // padded row stride (floats): 36*lid mod 64 -> 16 distinct banks

// C[M,N] = A[M,K] * B (+ bias[n]) * scale
//   NT==true : B is [N,K] row-major, used transposed (projections, qp @ kp^T)
//   NT==false: B is [K,N] row-major (att @ vp)
// Requires: M % 64 == 0, N % 128 == 0, K % 32 == 0.
template <bool NT, bool HAS_BIAS>
__global__ void __launch_bounds__(256)
wmma_gemm_f32(const float* __restrict__ A, const float* __restrict__ B,
              const float* __restrict__ bias, float* __restrict__ C,
              int M, int N, int K, float scale) {
  constexpr int BS_ELEMS = NT ? (128 * AS_STR) : (KC * 128);
  __shared__ float As[2][64 * AS_STR];
  __shared__ float Bs[2][BS_ELEMS];

  const int lane  = threadIdx.x & 31;
  const int wave  = threadIdx.x >> 5;
  const int wr    = wave >> 2;        // 0..1 -> 32 rows each
  const int wc    = wave & 3;         // 0..3 -> 32 cols each
  const int lid   = lane & 15;
  const int hi    = lane >> 4;        // lane half selects K pair for f32 frags
  const int kHalf = hi * 2;

  const int rowBlk = blockIdx.y * 64;
  const int colBlk = blockIdx.x * 128;
  const int n0 = colBlk + wc * 32 + lid;
  const int n1 = n0 + 16;

  // ---- async stage of one K chunk into LDS buffer `buf` (fully coalesced) ----
  const int sr  = threadIdx.x >> 3;        // 0..31
  const int sc4 = (threadIdx.x & 7) * 4;   // 0,4,...,28  (b128 column)
  auto stage = [&](int chunk, int buf) {
    const int kc = chunk * KC;
    // A tile: 64 rows x 32 k
#pragma unroll
    for (int rr = 0; rr < 64; rr += 32) {
      const float* g = A + (size_t)(rowBlk + sr + rr) * K + (kc + sc4);
      async_copy_b128(lds_addr_of(&As[buf][(sr + rr) * AS_STR + sc4]), g);
    }
    if (NT) {
      // B tile: 128 rows x 32 k, row-major with padded stride
#pragma unroll
      for (int rr = 0; rr < 128; rr += 32) {
        const float* g = B + (size_t)(colBlk + sr + rr) * K + (kc + sc4);
        async_copy_b128(lds_addr_of(&Bs[buf][(sr + rr) * AS_STR + sc4]), g);
      }
    } else {
      // B tile: 32 k-rows x 128 cols, k-major (consecutive lanes -> consecutive banks)
      const int c16 = (threadIdx.x & 7) * 16;
#pragma unroll
      for (int cc = 0; cc < 16; cc += 4) {
        const float* g = B + (size_t)(kc + sr) * N + (colBlk + c16 + cc);
        async_copy_b128(lds_addr_of(&Bs[buf][sr * 128 + c16 + cc]), g);
      }
    }
  };

  v8f acc00 = {}, acc01 = {}, acc10 = {}, acc11 = {};

  // ---- compute 8 k-steps (K=32) out of LDS buffer `buf` ----
  auto compute = [&](int buf) {
    const float* as = &As[buf][0];
    const float* bs = &Bs[buf][0];
#pragma unroll
    for (int s8 = 0; s8 < 8; ++s8) {
      const int kk = s8 * 4 + kHalf;
      v2f a0 = *(const v2f*)&as[(wr * 32 + lid) * AS_STR + kk];
      v2f a1 = *(const v2f*)&as[(wr * 32 + 16 + lid) * AS_STR + kk];
      v2f b0, b1;
      if (NT) {
        b0 = *(const v2f*)&bs[(wc * 32 + lid) * AS_STR + kk];
        b1 = *(const v2f*)&bs[(wc * 32 + 16 + lid) * AS_STR + kk];
      } else {
        b0.x = bs[(size_t)kk * 128 + wc * 32 + lid];
        b0.y = bs[(size_t)(kk + 1) * 128 + wc * 32 + lid];
        b1.x = bs[(size_t)kk * 128 + wc * 32 + 16 + lid];
        b1.y = bs[(size_t)(kk + 1) * 128 + wc * 32 + 16 + lid];
      }
      acc00 = WMMA_F32(a0, b0, acc00);
      acc01 = WMMA_F32(a0, b1, acc01);
      acc10 = WMMA_F32(a1, b0, acc10);
      acc11 = WMMA_F32(a1, b1, acc11);
    }
  };

  const int nchunks = K / KC;
  stage(0, 0);
  wait_async0();
  __syncthreads();
  for (int c = 0; c < nchunks; ++c) {
    const int buf = c & 1;
    if (c + 1 < nchunks) stage(c + 1, buf ^ 1);   // prefetch next chunk (async DMA)
    compute(buf);
    wait_async0();        // own async loads done ...
    __syncthreads();      // ... and every wave done reading current buffer
  }

  float bv0 = 0.0f, bv1 = 0.0f;
  if (HAS_BIAS) { bv0 = bias[n0]; bv1 = bias[n1]; }

  // C/D layout: VGPR r holds row (r + 8*hi), col = lane&15
  const int row0 = rowBlk + wr * 32 + hi * 8;
  const int row1 = row0 + 16;
#pragma unroll
  for (int r = 0; r < 8; ++r) {
    C[(size_t)(row0 + r) * N + n0] = (acc00[r] + bv0) * scale;
    C[(size_t)(row0 + r) * N + n1] = (acc01[r] + bv1) * scale;
    C[(size_t)(row1 + r) * N + n0] = (acc10[r] + bv0) * scale;
    C[(size_t)(row1 + r) * N + n1] = (acc11[r] + bv1) * scale;
  }
}

// Row softmax over N elements, then POST-softmax bias add (matches reference).
// One 256-thread block per row; scores updated in place.
__global__ void __launch_bounds__(256)
softmax_bias_kernel(float* __restrict__ scores, const float* __restrict__ bias, int N) {
  __shared__ float red[256];
  const int row = blockIdx.x;
  float* s = scores + (size_t)row * N;
  const float* b = bias + (size_t)row * N;

  float lmax = -3.402823466e38f;
  for (int j = threadIdx.x; j < N; j += 256) lmax = fmaxf(lmax, s[j]);
  red[threadIdx.x] = lmax;
  __syncthreads();
  for (int off = 128; off > 0; off >>= 1) {
    if (threadIdx.x < off)
      red[threadIdx.x] = fmaxf(red[threadIdx.x], red[threadIdx.x + off]);
    __syncthreads();
  }
  const float m = red[0];
  __syncthreads();

  float lsum = 0.0f;
  for (int j = threadIdx.x; j < N; j += 256) {
    float e = expf(s[j] - m);
    s[j] = e;
    lsum += e;
  }
  red[threadIdx.x] = lsum;
  __syncthreads();
  for (int off = 128; off > 0; off >>= 1) {
    if (threadIdx.x < off) red[threadIdx.x] += red[threadIdx.x + off];
    __syncthreads();
  }
  const float inv = 1.0f / red[0];
  for (int j = threadIdx.x; j < N; j += 256) s[j] = s[j] * inv + b[j];
}

extern "C" void kernel_launch(void* const* d_in, const int* in_sizes, int n_in,
                              void* d_out, int out_size, void* d_ws, size_t ws_size,
                              hipStream_t stream) {
  const float* q    = (const float*)d_in[0];
  const float* k    = (const float*)d_in[1];
  const float* v    = (const float*)d_in[2];
  const float* bias = (const float*)d_in[3];
  const float* Wq   = (const float*)d_in[4];
  const float* bq   = (const float*)d_in[5];
  const float* Wk   = (const float*)d_in[6];
  const float* bk   = (const float*)d_in[7];
  const float* Wv   = (const float*)d_in[8];
  const float* bv   = (const float*)d_in[9];

  const int D = in_sizes[5];           // 1024
  const int N = in_sizes[0] / D;       // 4096
  const float scale = 1.0f / sqrtf((float)D);

  float* qp = (float*)d_ws;
  float* kp = qp + (size_t)N * D;
  float* vp = kp + (size_t)N * D;
  float* sc = vp + (size_t)N * D;      // [N,N] scores/att, fits in 192MB L2

  dim3 blk(256);
  dim3 gproj(D / 128, N / 64);         // (8, 64)
  dim3 gsc(N / 128, N / 64);           // (32, 64)

  // qp = (q @ Wq^T + bq) * scale ; kp = k @ Wk^T + bk ; vp = v @ Wv^T + bv
  wmma_gemm_f32<true,  true ><<<gproj, blk, 0, stream>>>(q, Wq, bq, qp, N, D, D, scale);
  wmma_gemm_f32<true,  true ><<<gproj, blk, 0, stream>>>(k, Wk, bk, kp, N, D, D, 1.0f);
  wmma_gemm_f32<true,  true ><<<gproj, blk, 0, stream>>>(v, Wv, bv, vp, N, D, D, 1.0f);
  // scores = qp @ kp^T
  wmma_gemm_f32<true,  false><<<gsc,   blk, 0, stream>>>(qp, kp, nullptr, sc, N, N, D, 1.0f);
  // att = softmax(scores) + bias   (in place)
  softmax_bias_kernel<<<N, blk, 0, stream>>>(sc, bias, N);
  // out = att @ vp
  wmma_gemm_f32<false, false><<<gproj, blk, 0, stream>>>(sc, vp, nullptr, (float*)d_out,
                                                         N, D, N, 1.0f);
}